// StubLayer_11338713661687
// MI455X (gfx1250) — compile-verified
//
#include <hip/hip_runtime.h>
#include <hip/hip_bf16.h>

typedef __attribute__((ext_vector_type(2))) float v2f;
typedef __attribute__((ext_vector_type(8))) float v8f;

#define HIDDEN 2560
#define NH 8
#define HD 512
#define SEQ 32768
#define SCHUNK 128
#define NCHUNK (SEQ / SCHUNK)            /* 256 */
#define SCALE 0.04419417382415922f       /* 1/sqrt(512) */
#define NEG_INF (-__builtin_inff())

// workspace layout (float elements)
#define WS_Q 0
#define WS_K (WS_Q + NH * HD)                       // 4096
#define WS_V (WS_K + HD)                            // 4608
#define WS_M (WS_V + HD)                            // 5120
#define WS_L (WS_M + (NCHUNK + 1) * NH)
#define WS_ACC (WS_L + (NCHUNK + 1) * NH)
#define WS_ATTN (WS_ACC + (size_t)(NCHUNK + 1) * NH * HD)

#define WMMA_F32(a, b, c) \
  __builtin_amdgcn_wmma_f32_16x16x4_f32(false, (a), false, (b), (short)0, (c), false, false)

// ---------------- QKV projection: one wave per output row ----------------
__global__ __launch_bounds__(256) void qkv_gemv(const float* __restrict__ x,
                                                const float* __restrict__ wq,
                                                const float* __restrict__ wk,
                                                const float* __restrict__ wv,
                                                float* __restrict__ ws) {
  __shared__ float xs[HIDDEN];
  for (int i = threadIdx.x; i < HIDDEN; i += 256) xs[i] = x[i];
  __syncthreads();
  const int wave = threadIdx.x >> 5, lane = threadIdx.x & 31;
  const int row = blockIdx.x * 8 + wave;  // 0..5119
  const float* w;
  float* o;
  int r;
  if (row < NH * HD) {
    w = wq; r = row; o = ws + WS_Q + row;
  } else if (row < NH * HD + HD) {
    w = wk; r = row - NH * HD; o = ws + WS_K + r;
  } else {
    w = wv; r = row - NH * HD - HD; o = ws + WS_V + r;
  }
  const float4* w4 = (const float4*)(w + (size_t)r * HIDDEN);
  const float4* x4 = (const float4*)xs;
  float acc = 0.f;
#pragma unroll
  for (int i = 0; i < HIDDEN / 128; ++i) {
    float4 a = w4[i * 32 + lane];
    float4 b = x4[i * 32 + lane];
    acc += a.x * b.x + a.y * b.y + a.z * b.z + a.w * b.w;
  }
  for (int off = 16; off > 0; off >>= 1) acc += __shfl_down(acc, off, 32);
  if (lane == 0) *o = acc;
}

// ---------------- flash-decode chunk: f32 WMMA for q@K^T and P@V ----------------
__global__ __launch_bounds__(256) void attn_chunk(const float* __restrict__ kv_full,
                                                  const int* __restrict__ cur_pos_p,
                                                  float* __restrict__ ws) {
  __shared__ float qs[16][HD];       // 32 KB, heads padded to 16 (rows 8..15 zero)
  __shared__ float psm[16][SCHUNK];  //  8 KB, probs padded to 16 rows
  const int cur = *cur_pos_p;
  const float* Kc = kv_full;                     // [SEQ][HD]
  const float* Vc = kv_full + (size_t)SEQ * HD;  // [SEQ][HD]
  const float* qg = ws + WS_Q;
  const int base = blockIdx.x * SCHUNK;

  for (int i = threadIdx.x; i < 8 * HD; i += 256) {
    qs[i / HD][i % HD] = qg[i];
    qs[8 + i / HD][i % HD] = 0.f;
  }
  for (int i = threadIdx.x; i < 8 * SCHUNK; i += 256)
    psm[8 + i / SCHUNK][i % SCHUNK] = 0.f;
  __syncthreads();

  const int wave = threadIdx.x >> 5, lane = threadIdx.x & 31;
  const int mrow = lane & 15;       // M (A) and N (B) lane index
  const int ko = (lane >> 4) << 1;  // K offset for upper half-wave

  // ---- pass 1: one 16-head x 16-position score tile per wave ----
  {
    const int s0 = base + wave * 16;
    const float* krow = Kc + (size_t)(s0 + mrow) * HD + ko;  // B: N=mrow, K rows ko,ko+1
    const float* arow = &qs[mrow][ko];                       // A: M=mrow
    v8f c = {};
#pragma unroll 8
    for (int kk = 0; kk < HD; kk += 4) {
      v2f a = *(const v2f*)(arow + kk);
      v2f b = *(const v2f*)(krow + kk);
      c = WMMA_F32(a, b, c);
    }
    if (lane < 16) {
#pragma unroll
      for (int r = 0; r < 8; ++r) psm[r][wave * 16 + lane] = c[r] * SCALE;
    }
  }
  __syncthreads();

  // ---- softmax per head (wave w -> head w); mask stale cache slot at cur ----
  {
    const int h = wave;
    float mx = NEG_INF;
    for (int j = lane; j < SCHUNK; j += 32) {
      float sc = (base + j == cur) ? NEG_INF : psm[h][j];
      mx = fmaxf(mx, sc);
    }
#pragma unroll
    for (int o = 16; o > 0; o >>= 1) mx = fmaxf(mx, __shfl_xor(mx, o, 32));
    float sum = 0.f;
    for (int j = lane; j < SCHUNK; j += 32) {
      float p = (base + j == cur) ? 0.f : __expf(psm[h][j] - mx);
      psm[h][j] = p;
      sum += p;
    }
#pragma unroll
    for (int o = 16; o > 0; o >>= 1) sum += __shfl_xor(sum, o, 32);
    if (lane == 0) {
      ws[WS_M + blockIdx.x * NH + h] = mx;
      ws[WS_L + blockIdx.x * NH + h] = sum;
    }
  }
  __syncthreads();

  // ---- pass 2: 4 fused C-tiles (64 dims) per wave sharing one A fragment ----
  {
    const int d0 = wave * 64;
    const float* arow = &psm[mrow][ko];                            // A: P, M=head
    const float* vbase = Vc + (size_t)(base + ko) * HD + d0 + mrow;  // B: N=mrow
    v8f c0 = {}, c1 = {}, c2 = {}, c3 = {};
#pragma unroll 2
    for (int kp = 0; kp < SCHUNK; kp += 4) {
      v2f a = *(const v2f*)(arow + kp);
      const float* vp = vbase + (size_t)kp * HD;
      v2f b0, b1, b2, b3;
      b0.x = vp[0];  b0.y = vp[HD + 0];
      b1.x = vp[16]; b1.y = vp[HD + 16];
      b2.x = vp[32]; b2.y = vp[HD + 32];
      b3.x = vp[48]; b3.y = vp[HD + 48];
      c0 = WMMA_F32(a, b0, c0);
      c1 = WMMA_F32(a, b1, c1);
      c2 = WMMA_F32(a, b2, c2);
      c3 = WMMA_F32(a, b3, c3);
    }
    if (lane < 16) {
      float* accp = ws + WS_ACC + (size_t)blockIdx.x * NH * HD + d0 + lane;
#pragma unroll
      for (int r = 0; r < 8; ++r) {
        accp[(size_t)r * HD + 0] = c0[r];
        accp[(size_t)r * HD + 16] = c1[r];
        accp[(size_t)r * HD + 32] = c2[r];
        accp[(size_t)r * HD + 48] = c3[r];
      }
    }
  }
}

// ---------------- virtual chunk for the freshly-written K/V at cur_pos ----------------
__global__ __launch_bounds__(256) void curpos_partial(float* __restrict__ ws) {
  const float* q = ws + WS_Q;
  const float* kn = ws + WS_K;
  const float* vn = ws + WS_V;
  const int wave = threadIdx.x >> 5, lane = threadIdx.x & 31;
  {
    const int h = wave;
    float s = 0.f;
    for (int j = lane; j < HD; j += 32) s += q[h * HD + j] * kn[j];
#pragma unroll
    for (int o = 16; o > 0; o >>= 1) s += __shfl_xor(s, o, 32);
    if (lane == 0) {
      ws[WS_M + NCHUNK * NH + h] = s * SCALE;  // single element: l = exp(0) = 1
      ws[WS_L + NCHUNK * NH + h] = 1.0f;
    }
  }
  for (int idx = threadIdx.x; idx < NH * HD; idx += 256) {
    int h = idx / HD, d = idx % HD;
    ws[WS_ACC + ((size_t)NCHUNK * NH + h) * HD + d] = vn[d];
  }
}

// ---------------- merge partials (log-sum-exp) ----------------
__global__ __launch_bounds__(256) void combine(float* __restrict__ ws) {
  const int h = blockIdx.x;
  __shared__ float wls[NCHUNK + 1];
  float M = NEG_INF;
  for (int c = 0; c <= NCHUNK; ++c) M = fmaxf(M, ws[WS_M + c * NH + h]);
  for (int c = threadIdx.x; c <= NCHUNK; c += 256)
    wls[c] = __expf(ws[WS_M + c * NH + h] - M);
  __syncthreads();
  float den = 0.f;
  for (int c = 0; c <= NCHUNK; ++c) den += wls[c] * ws[WS_L + c * NH + h];
  const float inv = 1.0f / den;
  for (int d = threadIdx.x; d < HD; d += 256) {
    float num = 0.f;
    for (int c = 0; c <= NCHUNK; ++c)
      num += wls[c] * ws[WS_ACC + ((size_t)c * NH + h) * HD + d];
    ws[WS_ATTN + h * HD + d] = num * inv;
  }
}

// ---------------- output projection ----------------
__global__ __launch_bounds__(256) void out_gemv(const float* __restrict__ wo,
                                                const float* __restrict__ ws,
                                                float* __restrict__ out) {
  __shared__ float as[NH * HD];  // 16 KB
  for (int i = threadIdx.x; i < NH * HD; i += 256) as[i] = ws[WS_ATTN + i];
  __syncthreads();
  const int wave = threadIdx.x >> 5, lane = threadIdx.x & 31;
  const int row = blockIdx.x * 8 + wave;  // < 2560
  const float4* w4 = (const float4*)(wo + (size_t)row * (NH * HD));
  const float4* a4 = (const float4*)as;
  float acc = 0.f;
#pragma unroll
  for (int i = 0; i < (NH * HD) / 128; ++i) {
    float4 a = w4[i * 32 + lane];
    float4 b = a4[i * 32 + lane];
    acc += a.x * b.x + a.y * b.y + a.z * b.z + a.w * b.w;
  }
  for (int o = 16; o > 0; o >>= 1) acc += __shfl_down(acc, o, 32);
  if (lane == 0) out[row] = acc;
}

extern "C" void kernel_launch(void* const* d_in, const int* in_sizes, int n_in,
                              void* d_out, int out_size, void* d_ws, size_t ws_size,
                              hipStream_t stream) {
  const float* x = (const float*)d_in[0];
  // d_in[1] = kv_sliding (unused by the reference output)
  const float* kv_full = (const float*)d_in[2];
  const float* wq = (const float*)d_in[3];
  const float* wk = (const float*)d_in[4];
  const float* wv = (const float*)d_in[5];
  const float* wo = (const float*)d_in[6];
  const int* cur = (const int*)d_in[7];
  // d_in[8] = ring_pos (unused)
  float* ws = (float*)d_ws;
  float* out = (float*)d_out;

  qkv_gemv<<<(NH * HD + 2 * HD) / 8, 256, 0, stream>>>(x, wq, wk, wv, ws);
  attn_chunk<<<NCHUNK, 256, 0, stream>>>(kv_full, cur, ws);
  curpos_partial<<<1, 256, 0, stream>>>(ws);
  combine<<<NH, 256, 0, stream>>>(ws);
  out_gemv<<<HIDDEN / 8, 256, 0, stream>>>(wo, ws, out);
}